// CurveNet_SD_38208029065503
// MI455X (gfx1250) — compile-verified
//
// CurveNet backbone for MI455X (gfx1250), compile-only target.
//
// Design notes (MI455X roofline):
//  - Whole net is ~few GFLOP / ~150MB traffic for B=32 -> memory-latency bound
//    at 23.3 TB/s; so we fuse BN(g,b) + residual + leaky-ReLU into the GEMM
//    epilogue (one pass per tensor) and keep everything f32.
//  - All dense GEMMs go through V_WMMA_F32_16X16X4_F32 (one wave32 = one
//    16x16 f32 tile). Block = 4 waves covering a 16(O) x 64(N) tile; the
//    16xC weight tile is staged once in LDS (zero-padded, +2 stride pad),
//    so A-fragments are ds_load_b64 and B is unconditional global loads with
//    pointer-increment addressing. K unrolled x8 (2 WMMA/iter) + global
//    prefetch of the next B chunk. N-edge handled by a wave-uniform slow path
//    (EXEC stays all-ones at every WMMA, per ISA 7.12 restrictions).
//  - Irregular stages (kNN top-8, FPS, ball query, curve walk, softmaxes)
//    are tiny -> plain VALU kernels.
// Param mapping assumes the harness flattens the params pytree with jax's
// default tree order (dict keys sorted alphabetically at every level).

#include <hip/hip_runtime.h>
#include <cstdint>
#include <cstddef>

typedef __attribute__((ext_vector_type(2))) float v2f;
typedef __attribute__((ext_vector_type(8))) float v8f;

static inline int ceilDiv(int a, int b) { return (a + b - 1) / b; }

__device__ __forceinline__ float d_lrelu(float v) { return v > 0.f ? v : 0.2f * v; }

// ---------------------------------------------------------------------------
// WMMA GEMM:  Y[b,o,n] = act((sum_c W[o,c]*X[b,c,n]) * g[o] + bias[o] + R[b,o,n])
// grid = (ceil(N/64), ceil(O/16), B), block = 128 (4 waves).
// Requires C % 4 == 0 (true at every call site).
// act: 0 none, 1 lrelu(0.2), 2 relu
// ---------------------------------------------------------------------------
__global__ void k_gemm_wmma(const float* __restrict__ W, const float* __restrict__ X,
                            const float* __restrict__ gs, const float* __restrict__ bs,
                            const float* __restrict__ R, float* __restrict__ Y,
                            int O, int C, int N, int act)
{
    const int b    = blockIdx.z;
    const int o0   = blockIdx.y << 4;
    const int nblk = blockIdx.x << 6;          // 64 output columns per block
    const int wave = threadIdx.x >> 5;
    const int lane = threadIdx.x & 31;
    const int n0   = nblk + (wave << 4);
    const int row  = lane & 15;
    const int hi   = lane >> 4;                // 0 -> K{0,1}, 1 -> K{2,3}
    const int hi2  = hi << 1;
    const int strideA = C + 2;                 // LDS pad: dodge bank conflicts

    // Stage the 16 x C weight tile in LDS once per block (shared by 4 waves).
    extern __shared__ float As[];
    for (int i = threadIdx.x; i < 16 * C; i += 128) {
        int r = i / C, cc = i - r * C;
        As[r * strideA + cc] = (o0 + r < O) ? W[(size_t)(o0 + r) * C + cc] : 0.f;
    }
    __syncthreads();

    const float* Arow = As + row * strideA + hi2;
    const size_t bigN = (size_t)N;
    v8f acc = {0.f, 0.f, 0.f, 0.f, 0.f, 0.f, 0.f, 0.f};

    if (n0 + 16 <= N) {
        // Fast path: unconditional loads, pointer-increment addressing,
        // K unrolled x8 (two WMMAs in flight per iteration).
        const float* Bp = X + (size_t)b * C * N + (size_t)hi2 * N + (n0 + row);
        int k0 = 0;
        for (; k0 + 8 <= C; k0 += 8) {
            v2f a0 = *(const v2f*)(Arow + k0);
            v2f a1 = *(const v2f*)(Arow + k0 + 4);
            v2f b0; b0.x = Bp[0];        b0.y = Bp[bigN];
            v2f b1; b1.x = Bp[4 * bigN]; b1.y = Bp[5 * bigN];
            __builtin_prefetch(Bp + 8 * bigN, 0, 1);   // -> global_prefetch_b8
            acc = __builtin_amdgcn_wmma_f32_16x16x4_f32(false, a0, false, b0,
                                                        (short)0, acc, false, false);
            acc = __builtin_amdgcn_wmma_f32_16x16x4_f32(false, a1, false, b1,
                                                        (short)0, acc, false, false);
            Bp += 8 * bigN;
        }
        for (; k0 < C; k0 += 4) {
            v2f a0 = *(const v2f*)(Arow + k0);
            v2f b0; b0.x = Bp[0]; b0.y = Bp[bigN];
            acc = __builtin_amdgcn_wmma_f32_16x16x4_f32(false, a0, false, b0,
                                                        (short)0, acc, false, false);
            Bp += 4 * bigN;
        }
    } else {
        // N-edge path (wave-uniform branch; EXEC still all-ones at WMMA).
        const int bc = n0 + row;
        const bool inb = bc < N;
        const float* Bp = X + (size_t)b * C * N + (size_t)hi2 * N + (inb ? bc : 0);
        for (int k0 = 0; k0 < C; k0 += 4) {
            v2f a0 = *(const v2f*)(Arow + k0);
            v2f b0;
            b0.x = inb ? Bp[0]    : 0.f;
            b0.y = inb ? Bp[bigN] : 0.f;
            acc = __builtin_amdgcn_wmma_f32_16x16x4_f32(false, a0, false, b0,
                                                        (short)0, acc, false, false);
            Bp += 4 * bigN;
        }
    }

    const int ncol = n0 + row;
    if (ncol >= N) return;
    const int ob = o0 + hi * 8;
    float* Yb = Y + (size_t)b * O * N;
    const float* Rb = R ? (R + (size_t)b * O * N) : nullptr;
#pragma unroll
    for (int r = 0; r < 8; ++r) {
        int o = ob + r;
        if (o >= O) continue;
        float v = acc[r];
        if (gs) v = v * gs[o] + bs[o];
        if (Rb) v += Rb[(size_t)o * N + ncol];
        if (act == 1)      v = v > 0.f ? v : 0.2f * v;
        else if (act == 2) v = v > 0.f ? v : 0.f;
        Yb[(size_t)o * N + ncol] = v;
    }
}

// ---------------------------------------------------------------------------
// Misc small kernels
// ---------------------------------------------------------------------------
__global__ void k_transpose_pts(const float* __restrict__ xyz, float* __restrict__ pts,
                                int B, int N)
{   // (B,3,N) -> (B,N,3)
    int t = blockIdx.x * blockDim.x + threadIdx.x;
    if (t >= B * N) return;
    int b = t / N, n = t % N;
    const float* s = xyz + (size_t)b * 3 * N;
    float* d = pts + (size_t)t * 3;
    d[0] = s[n]; d[1] = s[N + n]; d[2] = s[2 * N + n];
}

__global__ void k_copy(float* __restrict__ dst, const float* __restrict__ src, int n)
{
    int t = blockIdx.x * blockDim.x + threadIdx.x;
    if (t < n) dst[t] = src[t];
}

// kNN: per (b,m) keep the 8 smallest squared distances (stable tie order).
__global__ void k_knn(const float* __restrict__ pts, int* __restrict__ idx, int B, int N)
{
    int t = blockIdx.x * blockDim.x + threadIdx.x;
    if (t >= B * N) return;
    int b = t / N, m = t % N;
    const float* P = pts + (size_t)b * N * 3;
    float mx = P[m * 3], my = P[m * 3 + 1], mz = P[m * 3 + 2];
    float bd[8]; int bi[8];
#pragma unroll
    for (int j = 0; j < 8; ++j) { bd[j] = 3.4e38f; bi[j] = 0; }
    for (int n = 0; n < N; ++n) {
        float dx = P[n * 3] - mx, dy = P[n * 3 + 1] - my, dz = P[n * 3 + 2] - mz;
        float d = dx * dx + dy * dy + dz * dz;
        if (d < bd[7]) {
            int j = 7;
            while (j > 0 && d < bd[j - 1]) { bd[j] = bd[j - 1]; bi[j] = bi[j - 1]; --j; }
            bd[j] = d; bi[j] = n;
        }
    }
#pragma unroll
    for (int j = 0; j < 8; ++j) idx[(size_t)t * 8 + j] = bi[j];
}

// FPS: one block per batch; matches the reference scan (emits `far` pre-update).
__global__ void k_fps(const float* __restrict__ pts, int* __restrict__ fidx,
                      int N, int npoint)
{
    int b = blockIdx.x, tid = threadIdx.x;
    const int T = 256;
    const float* P = pts + (size_t)b * N * 3;
    __shared__ float dist[1024];
    __shared__ float rv[256];
    __shared__ int   ri[256];
    for (int n = tid; n < N; n += T) dist[n] = 1e10f;
    __syncthreads();
    int far = 0;
    for (int it = 0; it < npoint; ++it) {
        if (tid == 0) fidx[b * npoint + it] = far;
        float cx = P[far * 3], cy = P[far * 3 + 1], cz = P[far * 3 + 2];
        float best = -1.f; int bidx = N;
        for (int n = tid; n < N; n += T) {
            float dx = P[n * 3] - cx, dy = P[n * 3 + 1] - cy, dz = P[n * 3 + 2] - cz;
            float d = dx * dx + dy * dy + dz * dz;
            float nd = fminf(dist[n], d);
            dist[n] = nd;
            if (nd > best) { best = nd; bidx = n; }
        }
        rv[tid] = best; ri[tid] = bidx;
        __syncthreads();
        for (int s = T / 2; s > 0; s >>= 1) {
            if (tid < s) {
                if (rv[tid + s] > rv[tid] ||
                    (rv[tid + s] == rv[tid] && ri[tid + s] < ri[tid])) {
                    rv[tid] = rv[tid + s]; ri[tid] = ri[tid + s];
                }
            }
            __syncthreads();
        }
        far = ri[0];
        __syncthreads();
    }
}

__global__ void k_gather_pts(const float* __restrict__ pts, const int* __restrict__ fidx,
                             float* __restrict__ out, int B, int N, int M)
{
    int t = blockIdx.x * blockDim.x + threadIdx.x;
    if (t >= B * M) return;
    int b = t / M, m = t % M;
    int j = fidx[b * M + m];
    const float* s = pts + ((size_t)b * N + j) * 3;
    float* d = out + (size_t)t * 3;
    d[0] = s[0]; d[1] = s[1]; d[2] = s[2];
}

// Ball query: first 8 indices (ascending) with d<=r^2, pad with first.
__global__ void k_ballquery(const float* __restrict__ pts, const float* __restrict__ np_,
                            int* __restrict__ gidx, int B, int N, int M, float r2)
{
    int t = blockIdx.x * blockDim.x + threadIdx.x;
    if (t >= B * M) return;
    int b = t / M, m = t % M;
    const float* P = pts + (size_t)b * N * 3;
    const float* q = np_ + (size_t)t * 3;
    float qx = q[0], qy = q[1], qz = q[2];
    int out[8]; int cnt = 0; int first = 0; bool have = false;
    for (int n = 0; n < N && cnt < 8; ++n) {
        float dx = P[n * 3] - qx, dy = P[n * 3 + 1] - qy, dz = P[n * 3 + 2] - qz;
        float d = dx * dx + dy * dy + dz * dz;
        if (d <= r2) { if (!have) { first = n; have = true; } out[cnt++] = n; }
    }
    for (; cnt < 8; ++cnt) out[cnt] = first;
#pragma unroll
    for (int k = 0; k < 8; ++k) gidx[(size_t)t * 8 + k] = out[k];
}

__global__ void k_pool_max(const float* __restrict__ x, const int* __restrict__ gidx,
                           float* __restrict__ y, int B, int C, int N, int M)
{
    int t = blockIdx.x * blockDim.x + threadIdx.x;
    if (t >= B * C * M) return;
    int m = t % M, c = (t / M) % C, b = t / (M * C);
    const float* xb = x + ((size_t)b * C + c) * N;
    const int* g = gidx + ((size_t)b * M + m) * 8;
    float best = -3.4e38f;
#pragma unroll
    for (int k = 0; k < 8; ++k) best = fmaxf(best, xb[g[k]]);
    y[((size_t)b * C + c) * M + m] = best;
}

// lpfa_initial: 9-dim geometric feature -> 32ch MLP -> lrelu -> max over k=8.
__global__ void k_lpfa_init(const float* __restrict__ pts, const int* __restrict__ knn,
                            const float* __restrict__ w, const float* __restrict__ g,
                            const float* __restrict__ bs, float* __restrict__ x,
                            int B, int N)
{
    int t = blockIdx.x * blockDim.x + threadIdx.x;
    if (t >= B * N * 32) return;
    int o = t % 32, n = (t / 32) % N, b = t / (32 * N);
    const float* P = pts + (size_t)b * N * 3;
    float cx = P[n * 3], cy = P[n * 3 + 1], cz = P[n * 3 + 2];
    float best = -3.4e38f;
    for (int k = 0; k < 8; ++k) {
        int j = knn[((size_t)b * N + n) * 8 + k];
        float nx = P[j * 3], ny = P[j * 3 + 1], nz = P[j * 3 + 2];
        float geo[9] = {cx, cy, cz, nx, ny, nz, nx - cx, ny - cy, nz - cz};
        float s = 0.f;
#pragma unroll
        for (int q = 0; q < 9; ++q) s += w[o * 9 + q] * geo[q];
        s = d_lrelu(s * g[o] + bs[o]);
        best = fmaxf(best, s);
    }
    x[((size_t)b * 32 + o) * N + n] = best;
}

// lpfa: f[b,c,n*4+k] = lrelu((x[c,idx]-x[c,n]) + bn(conv9(geo))[c])
__global__ void k_lpfa_f(const float* __restrict__ x, const float* __restrict__ pts,
                         const int* __restrict__ knn,
                         const float* __restrict__ xw, const float* __restrict__ xg,
                         const float* __restrict__ xb,
                         float* __restrict__ F, int B, int C, int N)
{
    int t = blockIdx.x * blockDim.x + threadIdx.x;
    if (t >= B * N * 4) return;
    int k = t & 3, n = (t >> 2) % N, b = t / (N * 4);
    int j = knn[((size_t)b * N + n) * 8 + k];
    const float* P = pts + (size_t)b * N * 3;
    float cx = P[n * 3], cy = P[n * 3 + 1], cz = P[n * 3 + 2];
    float nx = P[j * 3], ny = P[j * 3 + 1], nz = P[j * 3 + 2];
    float geo[9] = {cx, cy, cz, nx, ny, nz, nx - cx, ny - cy, nz - cz};
    const float* X = x + (size_t)b * C * N;
    float* Fb = F + (size_t)b * C * (N * 4);
    int col = n * 4 + k;
    for (int c = 0; c < C; ++c) {
        float p = 0.f;
#pragma unroll
        for (int q = 0; q < 9; ++q) p += xw[c * 9 + q] * geo[q];
        p = p * xg[c] + xb[c];
        float rel = X[(size_t)c * N + j] - X[(size_t)c * N + n];
        Fb[(size_t)c * (N * 4) + col] = d_lrelu(rel + p);
    }
}

__global__ void k_mean4(const float* __restrict__ hf, float* __restrict__ out,
                        int B, int C, int N)
{
    int t = blockIdx.x * blockDim.x + threadIdx.x;
    if (t >= B * C * N) return;
    int n = t % N; int bc = t / N;
    const float* s = hf + (size_t)bc * (N * 4) + n * 4;
    out[t] = 0.25f * (s[0] + s[1] + s[2] + s[3]);
}

// ---------------------------------------------------------------------------
// Curve grouping / aggregation
// ---------------------------------------------------------------------------
__global__ void k_att(const float* __restrict__ h, const float* __restrict__ aw,
                      float* __restrict__ att, int C, int N, int B)
{
    int t = blockIdx.x * blockDim.x + threadIdx.x;
    if (t >= B * N) return;
    int b = t / N, n = t % N;
    const float* X = h + (size_t)b * C * N;
    float s = 0.f;
    for (int c = 0; c < C; ++c) s += aw[c] * X[(size_t)c * N + n];
    att[t] = 1.f / (1.f + __expf(-s));
}

__global__ void k_scale(const float* __restrict__ h, const float* __restrict__ att,
                        float* __restrict__ y, int C, int N, int B)
{
    int t = blockIdx.x * blockDim.x + threadIdx.x;
    if (t >= B * C * N) return;
    int n = t % N; int b = t / (C * N);
    y[t] = h[t] * att[(size_t)b * N + n];
}

__global__ void k_top100(const float* __restrict__ att, int* __restrict__ start, int N)
{
    int b = blockIdx.x, tid = threadIdx.x;
    const int T = 256;
    __shared__ float a[1024];
    __shared__ float rv[256];
    __shared__ int   ri[256];
    for (int n = tid; n < N; n += T) a[n] = att[(size_t)b * N + n];
    __syncthreads();
    for (int it = 0; it < 100; ++it) {
        float best = -3.4e38f; int bi = 0x7fffffff;
        for (int n = tid; n < N; n += T)
            if (a[n] > best) { best = a[n]; bi = n; }
        rv[tid] = best; ri[tid] = bi;
        __syncthreads();
        for (int s = T / 2; s > 0; s >>= 1) {
            if (tid < s) {
                if (rv[tid + s] > rv[tid] ||
                    (rv[tid + s] == rv[tid] && ri[tid + s] < ri[tid])) {
                    rv[tid] = rv[tid + s]; ri[tid] = ri[tid + s];
                }
            }
            __syncthreads();
        }
        if (tid == 0) { start[b * 100 + it] = ri[0]; a[ri[0]] = -3.4e38f; }
        __syncthreads();
    }
}

// Walk: 5 sequential steps per (b,m) curve; C <= 32.
__global__ void k_walk(const float* __restrict__ xatt, const int* __restrict__ knn,
                       const int* __restrict__ start,
                       const float* __restrict__ agw, const float* __restrict__ agg,
                       const float* __restrict__ agb,
                       const float* __restrict__ mw, const float* __restrict__ mg,
                       const float* __restrict__ mb,
                       float* __restrict__ curves, int B, int C, int N)
{
    int t = blockIdx.x * blockDim.x + threadIdx.x;
    if (t >= B * 100) return;
    int b = t / 100, m = t % 100;
    const float* X = xatt + (size_t)b * C * N;
    float cur[32], pk[32];
    int ci = start[t];
    for (int c = 0; c < C; ++c) cur[c] = X[(size_t)c * N + ci];
    for (int step = 0; step < 5; ++step) {
        float bestS = -3.4e38f; int pi = ci;
        for (int k = 0; k < 7; ++k) {
            int nb = knn[((size_t)b * N + ci) * 8 + (k + 1)];
            float s = 0.f;
            for (int c = 0; c < C; ++c) s += agw[c] * X[(size_t)c * N + nb];
            for (int c = 0; c < C; ++c) s += agw[C + c] * cur[c];
            s = s * agg[0] + agb[0];
            if (s > bestS) { bestS = s; pi = nb; }
        }
        for (int c = 0; c < C; ++c) pk[c] = X[(size_t)c * N + pi];
        float s0 = 0.f, s1 = 0.f;
        for (int c = 0; c < C; ++c) {
            s0 += mw[c] * cur[c] + mw[C + c] * pk[c];
            s1 += mw[2 * C + c] * cur[c] + mw[3 * C + c] * pk[c];
        }
        s0 = s0 * mg[0] + mb[0];
        s1 = s1 * mg[1] + mb[1];
        float mxs = fmaxf(s0, s1);
        float e0 = __expf(s0 - mxs), e1 = __expf(s1 - mxs);
        float inv = 1.f / (e0 + e1);
        float w0 = e0 * inv, w1 = e1 * inv;
        for (int c = 0; c < C; ++c) {
            cur[c] = w0 * cur[c] + w1 * pk[c];
            curves[(((size_t)b * C + c) * 100 + m) * 5 + step] = cur[c];
        }
        ci = pi;
    }
}

// Curve attention reductions: inter (C,100) / intra (C,5) per batch.
__global__ void k_curve_reduce(const float* __restrict__ curves,
                               const float* __restrict__ lca,
                               float* __restrict__ inter, float* __restrict__ intra, int C)
{
    int b = blockIdx.x, tid = threadIdx.x;
    const int T = 256;
    __shared__ float catt[500], wl[500], wm[500];
    const float* Cv = curves + (size_t)b * C * 500;
    for (int i = tid; i < 500; i += T) {
        int m = i / 5, l = i % 5;
        float s = 0.f;
        for (int c = 0; c < C; ++c) s += lca[c] * Cv[(size_t)c * 500 + m * 5 + l];
        catt[i] = s;
    }
    __syncthreads();
    for (int m = tid; m < 100; m += T) {
        float mx = -3.4e38f;
        for (int l = 0; l < 5; ++l) mx = fmaxf(mx, catt[m * 5 + l]);
        float sum = 0.f;
        for (int l = 0; l < 5; ++l) { float e = __expf(catt[m * 5 + l] - mx); wl[m * 5 + l] = e; sum += e; }
        float inv = 1.f / sum;
        for (int l = 0; l < 5; ++l) wl[m * 5 + l] *= inv;
    }
    for (int l = tid; l < 5; l += T) {
        float mx = -3.4e38f;
        for (int m = 0; m < 100; ++m) mx = fmaxf(mx, catt[m * 5 + l]);
        float sum = 0.f;
        for (int m = 0; m < 100; ++m) { float e = __expf(catt[m * 5 + l] - mx); wm[m * 5 + l] = e; sum += e; }
        float inv = 1.f / sum;
        for (int m = 0; m < 100; ++m) wm[m * 5 + l] *= inv;
    }
    __syncthreads();
    for (int i = tid; i < C * 100; i += T) {
        int c = i / 100, m = i % 100;
        float s = 0.f;
        for (int l = 0; l < 5; ++l) s += Cv[(size_t)c * 500 + m * 5 + l] * wl[m * 5 + l];
        inter[(size_t)b * C * 100 + i] = s;
    }
    for (int i = tid; i < C * 5; i += T) {
        int c = i / 5, l = i % 5;
        float s = 0.f;
        for (int m = 0; m < 100; ++m) s += Cv[(size_t)c * 500 + m * 5 + l] * wm[m * 5 + l];
        intra[(size_t)b * C * 5 + i] = s;
    }
}

// s[b,n,m] = sum_d X1[b,d,n] * X2[b,d,m]
__global__ void k_bgemm_tn(const float* __restrict__ X1, const float* __restrict__ X2,
                           float* __restrict__ S, int B, int D, int N, int M)
{
    int t = blockIdx.x * blockDim.x + threadIdx.x;
    if (t >= B * N * M) return;
    int m = t % M, n = (t / M) % N, b = t / (M * N);
    const float* A = X1 + (size_t)b * D * N;
    const float* Bm = X2 + (size_t)b * D * M;
    float s = 0.f;
    for (int d = 0; d < D; ++d) s += A[(size_t)d * N + n] * Bm[(size_t)d * M + m];
    S[t] = s;
}

__global__ void k_softmax_row(float* __restrict__ buf, int rows, int M)
{
    int t = blockIdx.x * blockDim.x + threadIdx.x;
    if (t >= rows) return;
    float* r = buf + (size_t)t * M;
    float mx = -3.4e38f;
    for (int m = 0; m < M; ++m) mx = fmaxf(mx, r[m]);
    float sum = 0.f;
    for (int m = 0; m < M; ++m) { float e = __expf(r[m] - mx); r[m] = e; sum += e; }
    float inv = 1.f / sum;
    for (int m = 0; m < M; ++m) r[m] *= inv;
}

// out[b,n,off+o] = sum_m w[b,n,m] * A[b,o,m]
__global__ void k_apply_w(const float* __restrict__ Wm, const float* __restrict__ A,
                          float* __restrict__ out, int B, int N, int M, int mid,
                          int W2, int off)
{
    int t = blockIdx.x * blockDim.x + threadIdx.x;
    if (t >= B * N * mid) return;
    int o = t % mid, n = (t / mid) % N, b = t / (mid * N);
    const float* w = Wm + ((size_t)b * N + n) * M;
    const float* a = A + ((size_t)b * mid + o) * M;
    float s = 0.f;
    for (int m = 0; m < M; ++m) s += w[m] * a[m];
    out[((size_t)b * N + n) * W2 + off + o] = s;
}

// h[b,c,n] = lrelu(h[b,c,n] + sum_j dw[c,j]*f[b,n,j])   (in place)
__global__ void k_convd(const float* __restrict__ dw, const float* __restrict__ fcat,
                        float* __restrict__ h, int B, int C, int N, int J)
{
    int t = blockIdx.x * blockDim.x + threadIdx.x;
    if (t >= B * C * N) return;
    int n = t % N, c = (t / N) % C, b = t / (N * C);
    const float* f = fcat + ((size_t)b * N + n) * J;
    float s = 0.f;
    for (int j = 0; j < J; ++j) s += dw[c * J + j] * f[j];
    h[t] = d_lrelu(h[t] + s);
}

// ---------------------------------------------------------------------------
// Heads and FC
// ---------------------------------------------------------------------------
// vec[b, o] = max_n relu(bn(conv)) ; vec[b, O+o] = mean_n relu(bn(conv))
__global__ void k_head_fused(const float* __restrict__ w, const float* __restrict__ g,
                             const float* __restrict__ bs, const float* __restrict__ x,
                             float* __restrict__ vec, int B, int O, int C, int N)
{
    int t = blockIdx.x * blockDim.x + threadIdx.x;
    if (t >= B * O) return;
    int b = t / O, o = t % O;
    const float* X = x + (size_t)b * C * N;
    const float* wr = w + (size_t)o * C;
    float mx = -3.4e38f, sm = 0.f;
    for (int n = 0; n < N; ++n) {
        float s = 0.f;
        for (int c = 0; c < C; ++c) s += wr[c] * X[(size_t)c * N + n];
        s = s * g[o] + bs[o];
        s = s > 0.f ? s : 0.f;
        mx = fmaxf(mx, s);
        sm += s;
    }
    vec[(size_t)b * (2 * O) + o] = mx;
    vec[(size_t)b * (2 * O) + O + o] = sm / (float)N;
}

__global__ void k_fc_h(const float* __restrict__ v, const float* __restrict__ w1,
                       const float* __restrict__ g, const float* __restrict__ bs,
                       float* __restrict__ h, int B, int Cin, int H)
{
    int t = blockIdx.x * blockDim.x + threadIdx.x;
    if (t >= B * H) return;
    int b = t / H, i = t % H;
    const float* vb = v + (size_t)b * Cin;
    const float* wr = w1 + (size_t)i * Cin;
    float s = 0.f;
    for (int c = 0; c < Cin; ++c) s += vb[c] * wr[c];
    s = s * g[i] + bs[i];
    h[t] = s > 0.f ? s : 0.f;
}

__global__ void k_fc_o(const float* __restrict__ h, const float* __restrict__ w2,
                       const float* __restrict__ b2, float* __restrict__ out,
                       int B, int H, int O)
{
    int t = blockIdx.x * blockDim.x + threadIdx.x;
    if (t >= B * O) return;
    int b = t / O, o = t % O;
    const float* hb = h + (size_t)b * H;
    const float* wr = w2 + (size_t)o * H;
    float s = b2[o];
    for (int i = 0; i < H; ++i) s += hb[i] * wr[i];
    out[(size_t)b * O + o] = s;
}

// ---------------------------------------------------------------------------
// Host-side orchestration
// ---------------------------------------------------------------------------
namespace {

struct Conv { const float *b, *g, *w; };

struct Cursor {
    void* const* d; int i;
    const float* f() { return (const float*)d[i++]; }
};
static Conv loadConv(Cursor& c) { Conv v; v.b = c.f(); v.g = c.f(); v.w = c.f(); return v; }

struct CicP {
    Conv c1, c2, lmlp, lxyz, sc, agent, mom;
    const float *att_w, *lca_w, *conva, *convb, *convc, *convd, *convl, *convn;
    int cin, cout, pl, mid;
    bool has_sc, curve;
};

static CicP loadCic(Cursor& c, int cin, int cout, int ratio, bool curve) {
    CicP P{};
    P.cin = cin; P.cout = cout; P.pl = cin / ratio; P.mid = P.pl / 2;
    P.curve = curve; P.has_sc = (cin != cout);
    if (curve) {
        P.agent = loadConv(c);            // agent (b,g,w)
        P.att_w = c.f();                  // att_w
        P.c1 = loadConv(c);               // c1
        P.c2 = loadConv(c);               // c2
        P.conva = c.f(); P.convb = c.f(); P.convc = c.f();
        P.convd = c.f(); P.convl = c.f(); P.convn = c.f();
        P.lca_w = c.f();
        P.lmlp = loadConv(c);             // lpfa.mlp
        P.lxyz = loadConv(c);             // lpfa.xyz
        P.mom = loadConv(c);
        if (P.has_sc) P.sc = loadConv(c);
    } else {
        P.c1 = loadConv(c);
        P.c2 = loadConv(c);
        P.lmlp = loadConv(c);
        P.lxyz = loadConv(c);
        if (P.has_sc) P.sc = loadConv(c);
    }
    return P;
}

struct FcP { const float *b, *b2, *g, *w1, *w2; };
static FcP loadFc(Cursor& c) {
    FcP f; f.b = c.f(); f.b2 = c.f(); f.g = c.f(); f.w1 = c.f(); f.w2 = c.f(); return f;
}

struct Alloc {
    char* p;
    float* f(size_t n) { float* r = (float*)p; p += ((n * 4) + 255) & ~(size_t)255; return r; }
    int*   i(size_t n) { int*   r = (int*)p;   p += ((n * 4) + 255) & ~(size_t)255; return r; }
};

struct WS {
    float *ptsA, *ptsB, *XA, *XB, *SC, *H, *H2, *XATT, *F, *HF, *ATT;
    int *KNN, *FPS, *GIDX, *START;
    float *CURVES, *INTER, *INTRA, *Amat, *Bmat, *XC, *AN, *BL, *SINT, *SINTR, *FCAT;
    float *VEC, *HID;
};

static void gemm(const float* W, const float* X, const float* g, const float* b,
                 const float* R, float* Y, int B, int O, int C, int N, int act,
                 hipStream_t s)
{
    dim3 grid(ceilDiv(N, 64), ceilDiv(O, 16), B);
    size_t shmem = (size_t)16 * (C + 2) * sizeof(float);
    k_gemm_wmma<<<grid, 128, shmem, s>>>(W, X, g, b, R, Y, O, C, N, act);
}

static void swapf(float*& a, float*& b) { float* t = a; a = b; b = t; }

static void run_cic(const CicP& P, WS& w, float*& pts, float*& ptsAlt,
                    float*& x, float*& xAlt, int& N, int npoint, float radius,
                    int B, hipStream_t s)
{
    const int T = 256;
    if (N != npoint) {
        k_fps<<<B, 256, 0, s>>>(pts, w.FPS, N, npoint);
        k_gather_pts<<<ceilDiv(B * npoint, T), T, 0, s>>>(pts, w.FPS, ptsAlt, B, N, npoint);
        k_ballquery<<<ceilDiv(B * npoint, T), T, 0, s>>>(pts, ptsAlt, w.GIDX, B, N, npoint,
                                                         radius * radius);
        k_pool_max<<<ceilDiv(B * P.cin * npoint, T), T, 0, s>>>(x, w.GIDX, xAlt, B, P.cin,
                                                                N, npoint);
        swapf(pts, ptsAlt);
        swapf(x, xAlt);
        N = npoint;
    }
    const float* scp;
    if (P.has_sc) {
        gemm(P.sc.w, x, P.sc.g, P.sc.b, nullptr, w.SC, B, P.cout, P.cin, N, 0, s);
        scp = w.SC;
    } else {
        scp = x;
    }
    // h = lrelu(bn(c1(x)))
    gemm(P.c1.w, x, P.c1.g, P.c1.b, nullptr, w.H, B, P.pl, P.cin, N, 1, s);
    // kNN on current points
    k_knn<<<ceilDiv(B * N, T), T, 0, s>>>(pts, w.KNN, B, N);

    if (P.curve) {
        int C = P.pl, mid = P.mid;
        k_att<<<ceilDiv(B * N, T), T, 0, s>>>(w.H, P.att_w, w.ATT, C, N, B);
        k_scale<<<ceilDiv(B * C * N, T), T, 0, s>>>(w.H, w.ATT, w.XATT, C, N, B);
        k_top100<<<B, 256, 0, s>>>(w.ATT, w.START, N);
        k_walk<<<ceilDiv(B * 100, 128), 128, 0, s>>>(w.XATT, w.KNN, w.START,
                                                     P.agent.w, P.agent.g, P.agent.b,
                                                     P.mom.w, P.mom.g, P.mom.b,
                                                     w.CURVES, B, C, N);
        k_curve_reduce<<<B, 256, 0, s>>>(w.CURVES, P.lca_w, w.INTER, w.INTRA, C);
        gemm(P.conva, w.INTER, nullptr, nullptr, nullptr, w.Amat, B, mid, C, 100, 0, s);
        gemm(P.convb, w.INTRA, nullptr, nullptr, nullptr, w.Bmat, B, mid, C, 5, 0, s);
        gemm(P.convc, w.H, nullptr, nullptr, nullptr, w.XC, B, mid, C, N, 0, s);
        gemm(P.convn, w.Amat, nullptr, nullptr, nullptr, w.AN, B, mid, mid, 100, 0, s);
        gemm(P.convl, w.Bmat, nullptr, nullptr, nullptr, w.BL, B, mid, mid, 5, 0, s);
        k_bgemm_tn<<<ceilDiv(B * N * 100, T), T, 0, s>>>(w.XC, w.Amat, w.SINT, B, mid, N, 100);
        k_softmax_row<<<ceilDiv(B * N, T), T, 0, s>>>(w.SINT, B * N, 100);
        k_bgemm_tn<<<ceilDiv(B * N * 5, T), T, 0, s>>>(w.XC, w.Bmat, w.SINTR, B, mid, N, 5);
        k_softmax_row<<<ceilDiv(B * N, T), T, 0, s>>>(w.SINTR, B * N, 5);
        k_apply_w<<<ceilDiv(B * N * mid, T), T, 0, s>>>(w.SINT, w.AN, w.FCAT, B, N, 100,
                                                        mid, 2 * mid, 0);
        k_apply_w<<<ceilDiv(B * N * mid, T), T, 0, s>>>(w.SINTR, w.BL, w.FCAT, B, N, 5,
                                                        mid, 2 * mid, mid);
        k_convd<<<ceilDiv(B * C * N, T), T, 0, s>>>(P.convd, w.FCAT, w.H, B, C, N, 2 * mid);
    }
    // lpfa: f -> mlp(WMMA) -> mean over k=4
    k_lpfa_f<<<ceilDiv(B * N * 4, T), T, 0, s>>>(w.H, pts, w.KNN,
                                                 P.lxyz.w, P.lxyz.g, P.lxyz.b,
                                                 w.F, B, P.pl, N);
    gemm(P.lmlp.w, w.F, P.lmlp.g, P.lmlp.b, nullptr, w.HF, B, P.pl, P.pl, N * 4, 1, s);
    k_mean4<<<ceilDiv(B * P.pl * N, T), T, 0, s>>>(w.HF, w.H2, B, P.pl, N);
    // out = lrelu(bn(c2(h)) + sc)   (fused in the WMMA epilogue)
    gemm(P.c2.w, w.H2, P.c2.g, P.c2.b, scp, xAlt, B, P.cout, P.pl, N, 1, s);
    swapf(x, xAlt);
}

} // namespace

extern "C" void kernel_launch(void* const* d_in, const int* in_sizes, int n_in,
                              void* d_out, int out_size, void* d_ws, size_t ws_size,
                              hipStream_t stream)
{
    (void)in_sizes; (void)n_in; (void)ws_size;
    const int B = 32;
    const int T = 256;
    const float* xyz = (const float*)d_in[0];
    float* out = (float*)d_out;

    Cursor cur{d_in, 1};
    CicP cic11 = loadCic(cur, 32, 64, 2, true);
    CicP cic12 = loadCic(cur, 64, 64, 4, true);
    CicP cic21 = loadCic(cur, 64, 128, 2, true);
    CicP cic22 = loadCic(cur, 128, 128, 4, true);
    CicP cic31 = loadCic(cur, 128, 256, 2, false);
    CicP cic32 = loadCic(cur, 256, 256, 4, false);
    CicP cic41 = loadCic(cur, 256, 512, 2, false);
    CicP cic42 = loadCic(cur, 512, 512, 4, false);
    Conv conv01 = loadConv(cur);
    Conv conv02 = loadConv(cur);
    Conv conv03 = loadConv(cur);
    FcP fc1 = loadFc(cur);
    FcP fc2 = loadFc(cur);
    FcP fc3 = loadFc(cur);
    Conv lpfa0 = loadConv(cur);           // lpfa0.mlp

    // Workspace (bump-allocated; worst-case ~148 MB)
    Alloc al{(char*)d_ws};
    WS w;
    w.ptsA = al.f((size_t)B * 1024 * 3);
    w.ptsB = al.f((size_t)B * 1024 * 3);
    w.XA   = al.f((size_t)B * 128 * 1024);
    w.XB   = al.f((size_t)B * 128 * 1024);
    w.SC   = al.f((size_t)B * 128 * 1024);
    w.H    = al.f((size_t)B * 128 * 1024);
    w.H2   = al.f((size_t)B * 128 * 1024);
    w.XATT = al.f((size_t)B * 32 * 1024);
    w.F    = al.f((size_t)B * 32 * 1024 * 4);
    w.HF   = al.f((size_t)B * 32 * 1024 * 4);
    w.ATT  = al.f((size_t)B * 1024);
    w.KNN  = al.i((size_t)B * 1024 * 8);
    w.FPS  = al.i((size_t)B * 1024);
    w.GIDX = al.i((size_t)B * 1024 * 8);
    w.START = al.i((size_t)B * 100);
    w.CURVES = al.f((size_t)B * 32 * 100 * 5);
    w.INTER  = al.f((size_t)B * 32 * 100);
    w.INTRA  = al.f((size_t)B * 32 * 5);
    w.Amat   = al.f((size_t)B * 16 * 100);
    w.Bmat   = al.f((size_t)B * 16 * 5);
    w.XC     = al.f((size_t)B * 16 * 1024);
    w.AN     = al.f((size_t)B * 16 * 100);
    w.BL     = al.f((size_t)B * 16 * 5);
    w.SINT   = al.f((size_t)B * 1024 * 100);
    w.SINTR  = al.f((size_t)B * 1024 * 5);
    w.FCAT   = al.f((size_t)B * 1024 * 32);
    w.VEC    = al.f((size_t)B * 2048);
    w.HID    = al.f((size_t)B * 256);

    // ---- backbone ----
    int N = 1024;
    float *pts = w.ptsA, *ptsAlt = w.ptsB;
    float *x = w.XA, *xAlt = w.XB;

    k_transpose_pts<<<ceilDiv(B * N, T), T, 0, stream>>>(xyz, pts, B, N);
    k_knn<<<ceilDiv(B * N, T), T, 0, stream>>>(pts, w.KNN, B, N);
    k_lpfa_init<<<ceilDiv(B * N * 32, T), T, 0, stream>>>(pts, w.KNN, lpfa0.w, lpfa0.g,
                                                          lpfa0.b, x, B, N);

    run_cic(cic11, w, pts, ptsAlt, x, xAlt, N, 1024, 0.05f, B, stream);
    run_cic(cic12, w, pts, ptsAlt, x, xAlt, N, 1024, 0.05f, B, stream);
    run_cic(cic21, w, pts, ptsAlt, x, xAlt, N, 1024, 0.05f, B, stream);
    run_cic(cic22, w, pts, ptsAlt, x, xAlt, N, 1024, 0.10f, B, stream);

    // l2 = x (B,128,1024) -> out f2 ; head conv01 -> fc1 -> l1
    const size_t off_l1 = 0, off_l2 = 1280, off_l3 = 2560;
    const size_t off_f2 = 3840;
    const size_t off_f3 = off_f2 + (size_t)B * 128 * 1024;
    const size_t off_f4 = off_f3 + (size_t)B * 256 * 256;
    (void)out_size;
    k_copy<<<ceilDiv(B * 128 * 1024, T), T, 0, stream>>>(out + off_f2, x, B * 128 * 1024);
    k_head_fused<<<ceilDiv(B * 256, 128), 128, 0, stream>>>(conv01.w, conv01.g, conv01.b,
                                                            x, w.VEC, B, 256, 128, N);
    k_fc_h<<<ceilDiv(B * 64, T), T, 0, stream>>>(w.VEC, fc1.w1, fc1.g, fc1.b, w.HID,
                                                 B, 512, 64);
    k_fc_o<<<ceilDiv(B * 40, T), T, 0, stream>>>(w.HID, fc1.w2, fc1.b2, out + off_l1,
                                                 B, 64, 40);

    run_cic(cic31, w, pts, ptsAlt, x, xAlt, N, 256, 0.10f, B, stream);
    run_cic(cic32, w, pts, ptsAlt, x, xAlt, N, 256, 0.20f, B, stream);

    k_copy<<<ceilDiv(B * 256 * 256, T), T, 0, stream>>>(out + off_f3, x, B * 256 * 256);
    k_head_fused<<<ceilDiv(B * 512, 128), 128, 0, stream>>>(conv02.w, conv02.g, conv02.b,
                                                            x, w.VEC, B, 512, 256, N);
    k_fc_h<<<ceilDiv(B * 128, T), T, 0, stream>>>(w.VEC, fc2.w1, fc2.g, fc2.b, w.HID,
                                                  B, 1024, 128);
    k_fc_o<<<ceilDiv(B * 40, T), T, 0, stream>>>(w.HID, fc2.w2, fc2.b2, out + off_l2,
                                                 B, 128, 40);

    run_cic(cic41, w, pts, ptsAlt, x, xAlt, N, 64, 0.20f, B, stream);
    run_cic(cic42, w, pts, ptsAlt, x, xAlt, N, 64, 0.40f, B, stream);

    k_copy<<<ceilDiv(B * 512 * 64, T), T, 0, stream>>>(out + off_f4, x, B * 512 * 64);
    k_head_fused<<<ceilDiv(B * 1024, 128), 128, 0, stream>>>(conv03.w, conv03.g, conv03.b,
                                                             x, w.VEC, B, 1024, 512, N);
    k_fc_h<<<ceilDiv(B * 256, T), T, 0, stream>>>(w.VEC, fc3.w1, fc3.g, fc3.b, w.HID,
                                                  B, 2048, 256);
    k_fc_o<<<ceilDiv(B * 40, T), T, 0, stream>>>(w.HID, fc3.w2, fc3.b2, out + off_l3,
                                                 B, 256, 40);
}